// _CateModel_73074573574609
// MI455X (gfx1250) — compile-verified
//
#include <hip/hip_runtime.h>
#include <math.h>

#define EPS 1e-10f
#define NUM_CATES 100000
#define MAX_NGH 16
#define MAX_SCENES 10
#define EMB_DIM 64
#define HDIM 128   // [row_sum | agg]

typedef __attribute__((ext_vector_type(2))) float v2f;
typedef __attribute__((ext_vector_type(4))) float v4f;
typedef __attribute__((ext_vector_type(8))) float v8f;

// ---------------------------------------------------------------------------
// Pass 1: h[j, 0:64] = sum_k scene_emb[cate_scene_pad[j,k]]
// 16 lanes per row, each lane owns 4 consecutive floats (b128 loads/stores).
// ---------------------------------------------------------------------------
__global__ void rowsum_kernel(const float* __restrict__ scene_emb,
                              const int* __restrict__ cate_scene_pad,
                              float* __restrict__ h) {
  int gid = blockIdx.x * blockDim.x + threadIdx.x;
  int row = gid >> 4;
  int t   = gid & 15;
  if (row >= NUM_CATES) return;
  v4f acc = {0.f, 0.f, 0.f, 0.f};
  const int* idxp = cate_scene_pad + row * MAX_SCENES;
#pragma unroll
  for (int k = 0; k < MAX_SCENES; ++k) {
    int idx = idxp[k];
    acc += *(const v4f*)(scene_emb + idx * EMB_DIM + 4 * t);
  }
  *(v4f*)(h + row * HDIM + 4 * t) = acc;
}

// ---------------------------------------------------------------------------
// Pass 2: attention over [self, 16 neighbors]; writes h[j, 64:128] = agg.
// wave32 = two 16-lane groups, one row per group; lane owns 4 dims.
// Reductions via __shfl_xor butterflies (masks 1..8 stay inside each half).
// All gather traffic (h rows + cate_emb rows) is L2-resident (~77 MB << 192 MB).
// ---------------------------------------------------------------------------
__global__ void attn_kernel(const float* __restrict__ cate_emb,
                            const int* __restrict__ c_cate_pad,
                            float* __restrict__ h) {
  int lane = threadIdx.x & 31;
  int wave = threadIdx.x >> 5;
  int half = lane >> 4;
  int t    = lane & 15;
  int row  = (blockIdx.x * (blockDim.x >> 5) + wave) * 2 + half;
  if (row >= NUM_CATES) return;

  v4f my = *(const v4f*)(h + row * HDIM + 4 * t);
  v4f sq = my * my;
  float ss = sq.x + sq.y + sq.z + sq.w;
#pragma unroll
  for (int s = 1; s < 16; s <<= 1) ss += __shfl_xor(ss, s, 32);
  float inv_left = 1.0f / (ss + EPS);

  const int* nghp = c_cate_pad + row * MAX_NGH;
  float esum = 0.f;
  v4f agg = {0.f, 0.f, 0.f, 0.f};
  for (int i = 0; i < 1 + MAX_NGH; ++i) {
    int idx = (i == 0) ? row : nghp[i - 1];
    float m = (idx < NUM_CATES - 1) ? 1.0f : 0.0f;
    v4f r  = *(const v4f*)(h + idx * HDIM + 4 * t);
    v4f pd = my * r;
    v4f pr = r * r;
    float dp = pd.x + pd.y + pd.z + pd.w;
    float rr = pr.x + pr.y + pr.z + pr.w;
#pragma unroll
    for (int s = 1; s < 16; s <<= 1) {
      dp += __shfl_xor(dp, s, 32);
      rr += __shfl_xor(rr, s, 32);
    }
    // sim = m * dot / ((m*ss+eps)*(rr+eps)); m in {0,1} -> m in numerator suffices
    float sim = m * dp * inv_left / (rr + EPS);
    float e = __expf(sim) * m;
    esum += e;
    v4f c = *(const v4f*)(cate_emb + idx * EMB_DIM + 4 * t);
    agg += c * e;
  }
  agg *= (1.0f / (esum + EPS));
  *(v4f*)(h + row * HDIM + EMB_DIM + 4 * t) = agg;
}

// ---------------------------------------------------------------------------
// Pass 3: out = elu(h @ W^T + b) via V_WMMA_F32_16X16X4_F32 (fp32-exact).
// One wave per 16-row tile; 4 accumulators cover all 64 output columns.
// A (16x4 of h) and B (4x16 = W^T tile) striping: lane l holds M/N = l%16,
// K = 2*(l/16)+{0,1}  -> contiguous float2 per lane from row-major storage.
// C/D: VGPR r <-> output row M = r + 8*(l/16), column N = l%16.
// ---------------------------------------------------------------------------
__global__ void gemm_elu_kernel(const float* __restrict__ h,
                                const float* __restrict__ W,   // [64,128] row-major
                                const float* __restrict__ bvec, // [64]
                                float* __restrict__ out) {     // [N,64]
  int lane = threadIdx.x & 31;
  int wave = threadIdx.x >> 5;
  int m  = lane & 15;
  int hi = lane >> 4;
  int tile = blockIdx.x * (blockDim.x >> 5) + wave;
  if (tile >= NUM_CATES / 16) return;   // whole wave exits together
  int rowbase = tile * 16;

  const float* ha = h + (rowbase + m) * HDIM + 2 * hi;
  const float* wb = W + m * HDIM + 2 * hi;

  v8f c0 = {0,0,0,0,0,0,0,0};
  v8f c1 = {0,0,0,0,0,0,0,0};
  v8f c2 = {0,0,0,0,0,0,0,0};
  v8f c3 = {0,0,0,0,0,0,0,0};

#pragma unroll 4
  for (int k0 = 0; k0 < HDIM; k0 += 4) {
    v2f a  = *(const v2f*)(ha + k0);
    v2f b0 = *(const v2f*)(wb + k0);
    v2f b1 = *(const v2f*)(wb + 16 * HDIM + k0);
    v2f b2 = *(const v2f*)(wb + 32 * HDIM + k0);
    v2f b3 = *(const v2f*)(wb + 48 * HDIM + k0);
    c0 = __builtin_amdgcn_wmma_f32_16x16x4_f32(false, a, false, b0, (short)0, c0, false, false);
    c1 = __builtin_amdgcn_wmma_f32_16x16x4_f32(false, a, false, b1, (short)0, c1, false, false);
    c2 = __builtin_amdgcn_wmma_f32_16x16x4_f32(false, a, false, b2, (short)0, c2, false, false);
    c3 = __builtin_amdgcn_wmma_f32_16x16x4_f32(false, a, false, b3, (short)0, c3, false, false);
  }

  float bias0 = bvec[m], bias1 = bvec[16 + m], bias2 = bvec[32 + m], bias3 = bvec[48 + m];
#pragma unroll
  for (int r = 0; r < 8; ++r) {
    int row = rowbase + r + 8 * hi;
    float* op = out + row * EMB_DIM;
    float v;
    v = c0[r] + bias0; op[m]      = v > 0.f ? v : __expf(v) - 1.f;
    v = c1[r] + bias1; op[16 + m] = v > 0.f ? v : __expf(v) - 1.f;
    v = c2[r] + bias2; op[32 + m] = v > 0.f ? v : __expf(v) - 1.f;
    v = c3[r] + bias3; op[48 + m] = v > 0.f ? v : __expf(v) - 1.f;
  }
}

extern "C" void kernel_launch(void* const* d_in, const int* in_sizes, int n_in,
                              void* d_out, int out_size, void* d_ws, size_t ws_size,
                              hipStream_t stream) {
  (void)in_sizes; (void)n_in; (void)out_size; (void)ws_size;
  const float* cate_emb       = (const float*)d_in[1];
  const float* scene_emb      = (const float*)d_in[2];
  const int*   cate_scene_pad = (const int*)d_in[3];
  const int*   c_cate_pad     = (const int*)d_in[4];
  const float* agg_W          = (const float*)d_in[5];
  const float* agg_b          = (const float*)d_in[6];
  float* out = (float*)d_out;
  float* h   = (float*)d_ws;  // [NUM_CATES, 128] = 51.2 MB scratch

  // Pass 1: 16 lanes/row -> 16 rows per 256-thread block
  int blocks1 = (NUM_CATES * 16 + 255) / 256;
  rowsum_kernel<<<blocks1, 256, 0, stream>>>(scene_emb, cate_scene_pad, h);

  // Pass 2: 2 rows per wave, 8 waves per block -> 16 rows/block
  int blocks2 = (NUM_CATES + 15) / 16;
  attn_kernel<<<blocks2, 256, 0, stream>>>(cate_emb, c_cate_pad, h);

  // Pass 3: one wave per 16-row tile, 4 waves per block
  int tiles = NUM_CATES / 16;          // N divisible by 16
  int blocks3 = (tiles + 3) / 4;
  gemm_elu_kernel<<<blocks3, 128, 0, stream>>>(h, agg_W, agg_b, out);
}